// Model_16466904613150
// MI455X (gfx1250) — compile-verified
//
#include <hip/hip_runtime.h>
#include <hip/hip_bf16.h>

// ---------------------------------------------------------------------------
// Model dims: B=32 S=512 P=96 E=321 D=256 C=2 L=3
// Padded channel-mixer dims (multiples of 64, zero-padded => exact math):
//   EPAD=384, FPAD=704 (2E=642), GPAD=1344 (4E=1284), OPAD=384
// Workspace: ~254 MB fp32 (map | XT | H | G | x_stack | y | padded Wc_*)
// ---------------------------------------------------------------------------

typedef float v2f __attribute__((ext_vector_type(2)));
typedef float v8f __attribute__((ext_vector_type(8)));

__device__ __forceinline__ v8f wmma_f32x4(v2f a, v2f b, v8f c) {
  // D(16x16,f32) = A(16x4,f32) x B(4x16,f32) + C
  return __builtin_amdgcn_wmma_f32_16x16x4_f32(false, a, false, b, (short)0, c,
                                               false, false);
}

__device__ __forceinline__ float siluf(float v) {
  return v / (1.0f + __expf(-v));
}

// CDNA5 async copy global->LDS (ASYNCcnt). LDS operand is the wave-relative
// byte offset = low 32 bits of the generic shared-aperture pointer.
__device__ __forceinline__ void async_ld_b128(float* lds, const float* g) {
  asm volatile("global_load_async_to_lds_b128 %0, %1, off"
               :: "v"((unsigned)(uintptr_t)lds), "v"(g)
               : "memory");
}
__device__ __forceinline__ void wait_async0() {
  asm volatile("s_wait_asynccnt 0x0" ::: "memory");
}

// ---------------------------------------------------------------------------
// Channel -> weight-slot mapping tables (from GROUPS / SINGLES)
// ---------------------------------------------------------------------------
__constant__ int c_gflat[246] = {
  3,5,10,13,22,23,24,26,33,35,40,44,47,49,51,53,56,60,61,62,63,65,66,67,74,77,79,80,
  34,38,7,72,41,73,45,14,15,48,78,18,19,55,58,286,31,
  75,12,68,
  32,21,
  88,89,90,91,93,94,95,100,101,102,108,109,110,135,136,137,138,139,140,141,142,143,
  144,145,148,149,150,151,152,153,154,155,156,157,158,159,160,161,162,163,164,165,
  166,167,168,169,170,171,172,173,174,175,176,177,179,180,181,183,184,185,186,187,
  188,189,190,191,192,193,194,195,196,197,198,199,200,201,202,203,204,205,206,208,
  209,210,211,212,213,214,215,216,217,218,219,220,221,222,223,224,225,226,227,228,
  229,230,231,232,233,234,235,236,237,238,239,240,241,242,243,244,246,247,248,249,
  250,251,252,253,254,255,256,257,258,259,260,261,262,263,264,265,266,267,268,269,
  271,272,273,274,275,276,277,278,279,280,281,282,283,284,285,287,288,289,290,291,
  292,293,294,295,296,297,300,301,302,303,304,305,306,307,308,309,311,312,313,314,
  315,316,
  123,124,126,103,
  120,117,119,
  129,132,319,
  147,270
};
__constant__ int c_goff[10] = {0,28,45,48,50,234,238,241,244,246};
__constant__ int c_sing[75] = {
  0,1,2,4,6,8,9,11,16,17,20,25,27,28,29,30,36,37,39,42,43,46,50,52,54,57,59,64,
  69,70,71,76,81,82,83,84,85,86,87,92,96,97,98,99,104,105,106,107,111,112,113,114,
  115,116,118,121,122,125,127,128,130,131,133,134,146,178,182,207,245,298,299,310,
  317,318,320
};

__global__ void buildmap(int* __restrict__ map) {
  int t = threadIdx.x;
  if (t < 246) {
    int j = 0;
    #pragma unroll
    for (int q = 0; q < 9; ++q)
      if (t >= c_goff[q + 1]) j = q + 1;
    map[c_gflat[t]] = j;                 // kind 0 (group) | idx j
  } else if (t >= 256 && t < 256 + 75) {
    int n = t - 256;
    map[c_sing[n]] = (1 << 16) | n;      // kind 1 (single) | idx n
  }
}

// ---------------------------------------------------------------------------
// prep: xt = x - x[:, -1]; write padded XT (16384 x 384) and x_stack[0] (B,E,S)
// ---------------------------------------------------------------------------
__global__ void prep(const float* __restrict__ x, float* __restrict__ XT,
                     float* __restrict__ xs0) {
  int i = blockIdx.x * blockDim.x + threadIdx.x;
  if (i >= 32 * 512 * 384) return;
  int e = i % 384;
  int bs = i / 384;
  int s = bs & 511;
  int b = bs >> 9;
  float v = 0.0f;
  if (e < 321) {
    v = x[((size_t)b * 512 + s) * 321 + e] -
        x[((size_t)b * 512 + 511) * 321 + e];
    xs0[((size_t)b * 321 + e) * 512 + s] = v;
  }
  XT[i] = v;
}

// ---------------------------------------------------------------------------
// padw: zero-pad weight matrix (N,K) -> (Np,Kp), grid.y = c
// ---------------------------------------------------------------------------
__global__ void padw(const float* __restrict__ src, float* __restrict__ dst,
                     int N, int K, int Np, int Kp) {
  size_t so = (size_t)blockIdx.y * N * K;
  size_t doff = (size_t)blockIdx.y * Np * Kp;
  int i = blockIdx.x * blockDim.x + threadIdx.x;
  if (i >= Np * Kp) return;
  int k = i % Kp, n = i / Kp;
  dst[doff + i] = (n < N && k < K) ? src[so + (size_t)n * K + k] : 0.0f;
}

// ---------------------------------------------------------------------------
// gemm_nt<EPI>: C(MxNp) = A(MxK,lda) * W(NpxK,ldb)^T   (64x64x64 tiles)
// Double-buffered async global->LDS staging (ASYNCcnt) overlapped with WMMA.
//   EPI 0: raw store   EPI 1: silu   EPI 2: silu(Res + acc) -> C
//   EPI 3: transposed scatter into x_stack (n < Elim)
// Dynamic LDS: 4 * 64 * 68 * 4 = 69632 bytes.
// ---------------------------------------------------------------------------
template <int EPI>
__global__ __launch_bounds__(256) void gemm_nt(
    const float* __restrict__ A, int lda, const float* __restrict__ W, int ldb,
    float* __restrict__ C, int ldc, const float* __restrict__ Res,
    float* __restrict__ xs, int K, int Elim) {
  constexpr int BLD = 68;  // LDS row stride (bank-spread, 16B-aligned rows)
  extern __shared__ float sm[];
  float* buf[2][2];
  buf[0][0] = sm;                 // As buffer 0
  buf[0][1] = sm + 64 * BLD;      // Bs buffer 0
  buf[1][0] = sm + 2 * 64 * BLD;  // As buffer 1
  buf[1][1] = sm + 3 * 64 * BLD;  // Bs buffer 1

  int m0 = blockIdx.y * 64, n0 = blockIdx.x * 64;
  int tid = threadIdx.x;
  int lane = tid & 31, wave = tid >> 5;
  int lane15 = lane & 15, kh = lane >> 4;
  int mt = wave >> 1;            // 0..3
  int nt0 = (wave & 1) * 2;      // 0 or 2
  int nt1 = nt0 + 1;
  v8f acc0 = {}, acc1 = {};

  auto issue = [&](int bsel, int k0) {
    #pragma unroll
    for (int i = 0; i < 4; ++i) {
      int idx4 = tid + i * 256;          // 1024 float4 per matrix
      int r = idx4 >> 4;
      int c4 = (idx4 & 15) * 4;
      async_ld_b128(buf[bsel][0] + r * BLD + c4,
                    A + (size_t)(m0 + r) * lda + k0 + c4);
      async_ld_b128(buf[bsel][1] + r * BLD + c4,
                    W + (size_t)(n0 + r) * ldb + k0 + c4);
    }
  };

  int nchunks = K >> 6;
  issue(0, 0);
  for (int ci = 0; ci < nchunks; ++ci) {
    int cur = ci & 1;
    wait_async0();        // my loads for buf[cur] have landed in LDS
    __syncthreads();      // everyone's landed; prev compute on buf[cur^1] done
    if (ci + 1 < nchunks) issue(cur ^ 1, (ci + 1) << 6);

    const float* ar = buf[cur][0] + (mt * 16 + lane15) * BLD + 2 * kh;
    const float* br0 = buf[cur][1] + (nt0 * 16 + lane15) * BLD + 2 * kh;
    const float* br1 = buf[cur][1] + (nt1 * 16 + lane15) * BLD + 2 * kh;
    #pragma unroll
    for (int kk = 0; kk < 64; kk += 4) {
      v2f a = *(const v2f*)(ar + kk);
      v2f b0 = *(const v2f*)(br0 + kk);
      v2f b1 = *(const v2f*)(br1 + kk);
      acc0 = wmma_f32x4(a, b0, acc0);
      acc1 = wmma_f32x4(a, b1, acc1);
    }
  }

  int mbase = m0 + mt * 16 + 8 * kh;
  int ne0 = n0 + nt0 * 16 + lane15;
  int ne1 = n0 + nt1 * 16 + lane15;
  #pragma unroll
  for (int r = 0; r < 8; ++r) {
    int mm = mbase + r;
    float v0 = acc0[r], v1 = acc1[r];
    if constexpr (EPI == 0) {
      C[(size_t)mm * ldc + ne0] = v0;
      C[(size_t)mm * ldc + ne1] = v1;
    } else if constexpr (EPI == 1) {
      C[(size_t)mm * ldc + ne0] = siluf(v0);
      C[(size_t)mm * ldc + ne1] = siluf(v1);
    } else if constexpr (EPI == 2) {
      C[(size_t)mm * ldc + ne0] = siluf(Res[(size_t)mm * ldc + ne0] + v0);
      C[(size_t)mm * ldc + ne1] = siluf(Res[(size_t)mm * ldc + ne1] + v1);
    } else {  // EPI == 3: x_stack[(b,e,s)] = v for e < Elim
      int b = mm >> 9, s = mm & 511;
      if (ne0 < Elim) xs[((size_t)b * Elim + ne0) * 512 + s] = v0;
      if (ne1 < Elim) xs[((size_t)b * Elim + ne1) * 512 + s] = v1;
    }
  }
}

// ---------------------------------------------------------------------------
// mlp_tasks: one block per (l, e) task. Full 32-row MLP in LDS.
//   h = X @ Win^T ; g = silu(h @ Wr1^T) ; h = silu(h + g @ Wr2^T)
//   y = h @ Wout^T
// ---------------------------------------------------------------------------
__global__ __launch_bounds__(256) void mlp_tasks(
    const float* __restrict__ xstack, const int* __restrict__ map,
    const float* __restrict__ Wg_in, const float* __restrict__ Wg_r1,
    const float* __restrict__ Wg_r2, const float* __restrict__ Wg_out,
    const float* __restrict__ Ws_in, const float* __restrict__ Ws_r1,
    const float* __restrict__ Ws_r2, const float* __restrict__ Ws_out,
    float* __restrict__ y) {
  constexpr int XLD = 516;  // stride for 32x512 tiles (bank-spread, 16B rows)
  constexpr int HLD = 260;  // stride for 32x256 tile
  extern __shared__ float smem[];
  float* Xs = smem;                 // 32*XLD, aliased by gs after stage 1
  float* hs = smem + 32 * XLD;      // 32*HLD
  float* gs = smem;                 // alias of Xs (X dead after stage 1)

  int task = blockIdx.x;
  int l = task / 321, e = task % 321;
  int tid = threadIdx.x;
  int lane = tid & 31, wave = tid >> 5;
  int lane15 = lane & 15, kh = lane >> 4;

  int mv = map[e];
  const float *Win, *Wr1, *Wr2, *Wout;
  if ((mv >> 16) == 0) {
    size_t t = (size_t)(l * 9 + (mv & 0xffff));
    Win = Wg_in + t * (256 * 512);  Wr1 = Wg_r1 + t * (512 * 256);
    Wr2 = Wg_r2 + t * (256 * 512);  Wout = Wg_out + t * (96 * 256);
  } else {
    size_t t = (size_t)(l * 75 + (mv & 0xffff));
    Win = Ws_in + t * (256 * 512);  Wr1 = Ws_r1 + t * (512 * 256);
    Wr2 = Ws_r2 + t * (256 * 512);  Wout = Ws_out + t * (96 * 256);
  }
  // Start the weight streams early (global_prefetch_b8).
  __builtin_prefetch(Win, 0, 1);
  __builtin_prefetch(Wr1, 0, 1);
  __builtin_prefetch(Wr2, 0, 1);
  __builtin_prefetch(Wout, 0, 1);

  // Load X rows async into LDS: X[b, :] = x_stack[l, b, e, 0:512]
  #pragma unroll
  for (int i = 0; i < 16; ++i) {
    int idx4 = tid + i * 256;        // 4096 float4
    int row = idx4 >> 7;
    int c4 = (idx4 & 127) * 4;
    async_ld_b128(Xs + row * XLD + c4,
                  xstack + (((size_t)l * 32 + row) * 321 + e) * 512 + c4);
  }
  wait_async0();
  __syncthreads();

  // Stage 1: hs(32x256) = X(32x512) @ Win(256x512)^T
  for (int tt = wave; tt < 32; tt += 8) {
    int mt = tt & 1, nt = tt >> 1;
    const float* ar = Xs + (mt * 16 + lane15) * XLD + 2 * kh;
    const float* br = Win + (size_t)(nt * 16 + lane15) * 512 + 2 * kh;
    v8f acc = {};
    #pragma unroll 8
    for (int k = 0; k < 512; k += 4)
      acc = wmma_f32x4(*(const v2f*)(ar + k), *(const v2f*)(br + k), acc);
    int m8 = mt * 16 + 8 * kh, n = nt * 16 + lane15;
    #pragma unroll
    for (int r = 0; r < 8; ++r) hs[(m8 + r) * HLD + n] = acc[r];
  }
  __syncthreads();

  // Stage 2: gs(32x512) = silu(hs @ Wr1(512x256)^T)   (gs overwrites dead Xs)
  for (int tt = wave; tt < 64; tt += 8) {
    int mt = tt & 1, nt = tt >> 1;
    const float* ar = hs + (mt * 16 + lane15) * HLD + 2 * kh;
    const float* br = Wr1 + (size_t)(nt * 16 + lane15) * 256 + 2 * kh;
    v8f acc = {};
    #pragma unroll 8
    for (int k = 0; k < 256; k += 4)
      acc = wmma_f32x4(*(const v2f*)(ar + k), *(const v2f*)(br + k), acc);
    int m8 = mt * 16 + 8 * kh, n = nt * 16 + lane15;
    #pragma unroll
    for (int r = 0; r < 8; ++r) gs[(m8 + r) * XLD + n] = siluf(acc[r]);
  }
  __syncthreads();

  // Stage 3: hs = silu(hs + gs @ Wr2(256x512)^T)
  for (int tt = wave; tt < 32; tt += 8) {
    int mt = tt & 1, nt = tt >> 1;
    const float* ar = gs + (mt * 16 + lane15) * XLD + 2 * kh;
    const float* br = Wr2 + (size_t)(nt * 16 + lane15) * 512 + 2 * kh;
    v8f acc = {};
    #pragma unroll 8
    for (int k = 0; k < 512; k += 4)
      acc = wmma_f32x4(*(const v2f*)(ar + k), *(const v2f*)(br + k), acc);
    int m8 = mt * 16 + 8 * kh, n = nt * 16 + lane15;
    #pragma unroll
    for (int r = 0; r < 8; ++r) {
      int o = (m8 + r) * HLD + n;
      hs[o] = siluf(hs[o] + acc[r]);   // each element owned by one thread
    }
  }
  __syncthreads();

  // Stage 4: y(32x96) = hs @ Wout(96x256)^T -> y[l, b, e, p]
  for (int tt = wave; tt < 12; tt += 8) {
    int mt = tt & 1, nt = tt >> 1;
    const float* ar = hs + (mt * 16 + lane15) * HLD + 2 * kh;
    const float* br = Wout + (size_t)(nt * 16 + lane15) * 256 + 2 * kh;
    v8f acc = {};
    #pragma unroll 8
    for (int k = 0; k < 256; k += 4)
      acc = wmma_f32x4(*(const v2f*)(ar + k), *(const v2f*)(br + k), acc);
    int m8 = mt * 16 + 8 * kh, n = nt * 16 + lane15;
    #pragma unroll
    for (int r = 0; r < 8; ++r)
      y[(((size_t)l * 32 + m8 + r) * 321 + e) * 96 + n] = acc[r];
  }
}

// ---------------------------------------------------------------------------
// combine: out[b,p,e] = 0.5*y0 + 0.25*(y1+y2) + x[b, S-1, e]
// ---------------------------------------------------------------------------
__global__ void combine(const float* __restrict__ y, const float* __restrict__ x,
                        float* __restrict__ out) {
  int i = blockIdx.x * blockDim.x + threadIdx.x;
  if (i >= 32 * 96 * 321) return;
  int e = i % 321;
  int bp = i / 321;
  int p = bp % 96;
  int b = bp / 96;
  const size_t YL = (size_t)32 * 321 * 96;
  size_t yi = (((size_t)b) * 321 + e) * 96 + p;
  float v = 0.5f * y[yi] + 0.25f * (y[yi + YL] + y[yi + 2 * YL]);
  out[i] = v + x[((size_t)b * 512 + 511) * 321 + e];
}

// ---------------------------------------------------------------------------
extern "C" void kernel_launch(void* const* d_in, const int* in_sizes, int n_in,
                              void* d_out, int out_size, void* d_ws,
                              size_t ws_size, hipStream_t stream) {
  (void)in_sizes; (void)n_in; (void)out_size; (void)ws_size;
  const float* x = (const float*)d_in[0];
  const float* Wg_in = (const float*)d_in[1];
  const float* Wg_r1 = (const float*)d_in[2];
  const float* Wg_r2 = (const float*)d_in[3];
  const float* Wg_out = (const float*)d_in[4];
  const float* Ws_in = (const float*)d_in[5];
  const float* Ws_r1 = (const float*)d_in[6];
  const float* Ws_r2 = (const float*)d_in[7];
  const float* Ws_out = (const float*)d_in[8];
  const float* Wc_in = (const float*)d_in[9];
  const float* Wc_r1 = (const float*)d_in[10];
  const float* Wc_r2 = (const float*)d_in[11];
  const float* Wc_out = (const float*)d_in[12];
  float* out = (float*)d_out;

  // Workspace layout (fp32, ~254 MB total)
  int* map = (int*)d_ws;
  float* XT = (float*)d_ws + 512;                    // 16384 x 384
  float* H = XT + (size_t)16384 * 384;               // 16384 x 704
  float* G = H + (size_t)16384 * 704;                // 16384 x 1344
  float* XS = G + (size_t)16384 * 1344;              // x_stack: 3 x 32 x 321 x 512
  float* Y = XS + (size_t)3 * 32 * 321 * 512;        // y: 3 x 32 x 321 x 96
  float* Wip = Y + (size_t)3 * 32 * 321 * 96;        // 2 x 704 x 384
  float* W1p = Wip + (size_t)2 * 704 * 384;          // 2 x 1344 x 704
  float* W2p = W1p + (size_t)2 * 1344 * 704;         // 2 x 704 x 1344
  float* Wop = W2p + (size_t)2 * 704 * 1344;         // 2 x 384 x 704

  buildmap<<<1, 352, 0, stream>>>(map);
  prep<<<(32 * 512 * 384 + 255) / 256, 256, 0, stream>>>(x, XT, XS);
  padw<<<dim3((704 * 384 + 255) / 256, 2), 256, 0, stream>>>(Wc_in, Wip, 642, 321, 704, 384);
  padw<<<dim3((1344 * 704 + 255) / 256, 2), 256, 0, stream>>>(Wc_r1, W1p, 1284, 642, 1344, 704);
  padw<<<dim3((704 * 1344 + 255) / 256, 2), 256, 0, stream>>>(Wc_r2, W2p, 642, 1284, 704, 1344);
  padw<<<dim3((384 * 704 + 255) / 256, 2), 256, 0, stream>>>(Wc_out, Wop, 321, 642, 384, 704);

  const unsigned gemmLds = 4u * 64 * 68 * sizeof(float);  // 69632 B
  for (int c = 0; c < 2; ++c) {
    // h = xt @ Wc_in^T
    gemm_nt<0><<<dim3(704 / 64, 16384 / 64), 256, gemmLds, stream>>>(
        XT, 384, Wip + (size_t)c * 704 * 384, 384, H, 704, nullptr, nullptr, 384, 0);
    // g = silu(h @ Wc_r1^T)
    gemm_nt<1><<<dim3(1344 / 64, 16384 / 64), 256, gemmLds, stream>>>(
        H, 704, W1p + (size_t)c * 1344 * 704, 704, G, 1344, nullptr, nullptr, 704, 0);
    // h = silu(h + g @ Wc_r2^T)
    gemm_nt<2><<<dim3(704 / 64, 16384 / 64), 256, gemmLds, stream>>>(
        G, 1344, W2p + (size_t)c * 704 * 1344, 1344, H, 704, H, nullptr, 1344, 0);
    // x_stack[1+c] = transpose(h @ Wc_out^T)
    gemm_nt<3><<<dim3(384 / 64, 16384 / 64), 256, gemmLds, stream>>>(
        H, 704, Wop + (size_t)c * 384 * 704, 704, nullptr, 0, nullptr,
        XS + (size_t)(1 + c) * 32 * 321 * 512, 704, 321);
  }

  // 963 (l,e) tasks; dynamic LDS: 32*516 + 32*260 floats = 99328 bytes
  mlp_tasks<<<963, 256, (32 * 516 + 32 * 260) * sizeof(float), stream>>>(
      XS, map, Wg_in, Wg_r1, Wg_r2, Wg_out, Ws_in, Ws_r1, Ws_r2, Ws_out, Y);

  combine<<<(32 * 96 * 321 + 255) / 256, 256, 0, stream>>>(Y, x, out);
}